// AVWDCRNN_76931454206159
// MI455X (gfx1250) — compile-verified
//
// AVWDCRNN (AGCRN) for MI455X / gfx1250 — fp32 WMMA (V_WMMA_F32_16X16X4_F32)
// with LDS-staged, double-buffered graph GEMM (async-to-LDS when available).
//
// Decomposition:
//   A = softmax(relu(E E^T))                       [k_scores]
//   Wn[n,k,i,o] = sum_d E[n,d] w_pool[d,k,i,o]     [k_mix] (time-invariant, done once)
//   per step: conv = A @ X (batched, WMMA)         [k_gemm_graph]
//             out  = act(X@Wn0 + conv@Wn1 + bn)    [k_gemm_node, WMMA]
//             GRU combine                          [k_gru]
//
// Workspace requirement: ~488 MB of float scratch.

#include <hip/hip_runtime.h>
#include <cstdint>
#include <cstddef>

typedef __attribute__((ext_vector_type(2))) float v2f;
typedef __attribute__((ext_vector_type(8))) float v8f;
typedef int v2i_vs __attribute__((vector_size(8)));   // matches builtin param type

#define N_NODES 1024
#define BATCH   32
#define TSTEPS  12
#define DEMB    10
#define HID     64
#define DIN     2
#define BK      64   // k-slab staged in LDS per chunk

// gfx1250 async global->LDS path (ASYNCcnt), with synchronous fallback.
#if defined(__has_builtin)
#if __has_builtin(__builtin_amdgcn_global_load_async_to_lds_b64) && \
    __has_builtin(__builtin_amdgcn_s_wait_asynccnt)
#define USE_ASYNC_LDS 1
#endif
#endif
#ifndef USE_ASYNC_LDS
#define USE_ASYNC_LDS 0
#endif

__device__ __forceinline__ void stage_b64(const float* gsrc, float* ldst) {
#if USE_ASYNC_LDS
  __builtin_amdgcn_global_load_async_to_lds_b64(
      (__attribute__((address_space(1))) v2i_vs*)gsrc,
      (__attribute__((address_space(3))) v2i_vs*)ldst, 0, 0);
#else
  *(v2f*)ldst = *(const v2f*)gsrc;
#endif
}

__device__ __forceinline__ void stage_wait() {
#if USE_ASYNC_LDS
  __builtin_amdgcn_s_wait_asynccnt(0);
#endif
}

__device__ __forceinline__ v8f wmma_f32(v2f a, v2f b, v8f c) {
  // D = A(16x4) * B(4x16) + C(16x16), fp32
  return __builtin_amdgcn_wmma_f32_16x16x4_f32(false, a, false, b, (short)0, c,
                                               false, false);
}

__device__ __forceinline__ float sigmoidf_(float x) {
  return 1.0f / (1.0f + __expf(-x));
}

// ---------------------------------------------------------------------------
// A = softmax(relu(E E^T), axis=1). One block per row, 256 threads, 4 cols each.
// ---------------------------------------------------------------------------
__global__ void k_scores(const float* __restrict__ E, float* __restrict__ Amat) {
  __shared__ float en[DEMB];
  __shared__ float red[256];
  const int n = blockIdx.x;
  const int tid = threadIdx.x;
  if (tid < DEMB) en[tid] = E[(size_t)n * DEMB + tid];
  __syncthreads();

  float vals[4];
  float mx = -1e30f;
#pragma unroll
  for (int j = 0; j < 4; ++j) {
    int m = tid + j * 256;
    float s = 0.f;
#pragma unroll
    for (int d = 0; d < DEMB; ++d) s += en[d] * E[(size_t)m * DEMB + d];
    s = fmaxf(s, 0.f);
    vals[j] = s;
    mx = fmaxf(mx, s);
  }
  red[tid] = mx;
  __syncthreads();
  for (int o = 128; o > 0; o >>= 1) {
    if (tid < o) red[tid] = fmaxf(red[tid], red[tid + o]);
    __syncthreads();
  }
  mx = red[0];
  __syncthreads();

  float sum = 0.f;
#pragma unroll
  for (int j = 0; j < 4; ++j) {
    vals[j] = __expf(vals[j] - mx);
    sum += vals[j];
  }
  red[tid] = sum;
  __syncthreads();
  for (int o = 128; o > 0; o >>= 1) {
    if (tid < o) red[tid] += red[tid + o];
    __syncthreads();
  }
  const float inv = 1.0f / red[0];
#pragma unroll
  for (int j = 0; j < 4; ++j)
    Amat[(size_t)n * N_NODES + tid + j * 256] = vals[j] * inv;
}

// ---------------------------------------------------------------------------
// Outp[n*L + j] = sum_d E[n,d] * Pool[d*L + j]   (per-node weights & biases)
// ---------------------------------------------------------------------------
__global__ void k_mix(const float* __restrict__ E, const float* __restrict__ Pool,
                      float* __restrict__ Outp, int L) {
  size_t idx = (size_t)blockIdx.x * blockDim.x + threadIdx.x;
  size_t total = (size_t)N_NODES * (size_t)L;
  if (idx >= total) return;
  int n = (int)(idx / L);
  int j = (int)(idx % L);
  float s = 0.f;
#pragma unroll
  for (int d = 0; d < DEMB; ++d)
    s += E[(size_t)n * DEMB + d] * Pool[(size_t)d * L + j];
  Outp[idx] = s;
}

// ---------------------------------------------------------------------------
// state(N,B,H) <- init_state[layer] (layout (2,B,N,H))
// ---------------------------------------------------------------------------
__global__ void k_load_state(const float* __restrict__ init, float* __restrict__ st,
                             int layer) {
  size_t idx = (size_t)blockIdx.x * blockDim.x + threadIdx.x;
  const size_t total = (size_t)N_NODES * BATCH * HID;
  if (idx >= total) return;
  int n = (int)(idx / (BATCH * HID));
  int r = (int)(idx % (BATCH * HID));
  int b = r / HID, o = r % HID;
  st[idx] = init[(((size_t)layer * BATCH + b) * N_NODES + n) * HID + o];
}

// ---------------------------------------------------------------------------
// Build inp/cand (N,B,C): [x_t | state] or [x_t | z*state]
// ---------------------------------------------------------------------------
__global__ void k_build(const float* __restrict__ x0, const float* __restrict__ h0seq,
                        const float* __restrict__ st, const float* __restrict__ zr,
                        float* __restrict__ dst, int layer, int t, int Cin, int use_z) {
  const int C = Cin + HID;
  size_t total = (size_t)N_NODES * BATCH * C;
  size_t idx = (size_t)blockIdx.x * blockDim.x + threadIdx.x;
  if (idx >= total) return;
  int n = (int)(idx / (BATCH * C));
  int rem = (int)(idx % (BATCH * C));
  int b = rem / C, c = rem % C;
  float v;
  if (c < Cin) {
    v = (layer == 0)
            ? x0[(((size_t)b * TSTEPS + t) * N_NODES + n) * DIN + c]
            : h0seq[(((size_t)t * N_NODES + n) * BATCH + b) * HID + c];
  } else {
    float sv = st[((size_t)n * BATCH + b) * HID + (c - Cin)];
    if (use_z) sv *= zr[((size_t)n * BATCH + b) * (2 * HID) + (c - Cin)];
    v = sv;
  }
  dst[idx] = v;
}

// ---------------------------------------------------------------------------
// Graph conv GEMM: Y[n,b,c] = sum_k A[n,k] * X[k,b,c]   (per batch b)
// X,Y node-major (N,B,C). Block = 8 waves = 128 rows x 16 cols, grid.z = batch.
// X k-slab (BK x 16) staged in LDS, double-buffered, shared by all 8 waves;
// transfers issued via GLOBAL_LOAD_ASYNC_TO_LDS_B64 when available.
// Edge columns (C=66) load in-allocation garbage; those output columns are
// masked on store, and D-matrix columns are independent, so no zeroing needed.
// ---------------------------------------------------------------------------
__global__ void k_gemm_graph(const float* __restrict__ Amat, const float* __restrict__ X,
                             float* __restrict__ Y, int C) {
  __shared__ float xt[2][BK * 16];
  const int tid = threadIdx.x;
  const int wave = tid >> 5;
  const int lane = tid & 31;
  const int row0 = blockIdx.x * 128 + wave * 16;
  const int col0 = blockIdx.y * 16;
  const int b = blockIdx.z;
  const int m = lane & 15;
  const int kb = (lane < 16) ? 0 : 2;
  const int nn = lane & 15;
  const int colv = col0 + nn;
  const bool cok = (colv < C);

  const float* __restrict__ arow = Amat + (size_t)(row0 + m) * N_NODES;

  // Staging map: BK*16 = 1024 floats = 512 b64 pairs; 256 threads x 2 pairs.
  // pair p -> k_local = p >> 3, col = (p & 7) * 2
  const int p0 = tid, p1 = tid + 256;
  const int kl0 = p0 >> 3, sc0 = (p0 & 7) * 2;
  const int kl1 = p1 >> 3, sc1 = (p1 & 7) * 2;
  const size_t gstep = (size_t)BATCH * C;  // between k rows of X
  const float* gbase = X + (size_t)b * C + col0;

  // prologue: stage chunk 0
  stage_b64(gbase + (size_t)kl0 * gstep + sc0, &xt[0][kl0 * 16 + sc0]);
  stage_b64(gbase + (size_t)kl1 * gstep + sc1, &xt[0][kl1 * 16 + sc1]);
  stage_wait();
  __syncthreads();

  v8f acc = {};
  const int NCH = N_NODES / BK;
  for (int ch = 0; ch < NCH; ++ch) {
    const int cur = ch & 1;
    if (ch + 1 < NCH) {
      const float* g = gbase + (size_t)(ch + 1) * BK * gstep;
      stage_b64(g + (size_t)kl0 * gstep + sc0, &xt[cur ^ 1][kl0 * 16 + sc0]);
      stage_b64(g + (size_t)kl1 * gstep + sc1, &xt[cur ^ 1][kl1 * 16 + sc1]);
      // next A-row slab: high temporal locality -> WGP-scope prefetch
      __builtin_prefetch(arow + (ch + 1) * BK, 0, 3);
    }
    const float* xs = &xt[cur][0];
    const int kk0 = ch * BK;
#pragma unroll
    for (int kl = 0; kl < BK; kl += 4) {
      v2f a = *(const v2f*)(arow + kk0 + kl + kb);  // contiguous A[row][k..k+1]
      v2f bv;
      bv.x = xs[(kl + kb) * 16 + nn];
      bv.y = xs[(kl + kb + 1) * 16 + nn];
      acc = wmma_f32(a, bv, acc);
    }
    stage_wait();
    __syncthreads();
  }

  if (cok) {
    const int mbase = row0 + ((lane < 16) ? 0 : 8);
#pragma unroll
    for (int v = 0; v < 8; ++v)
      Y[((size_t)(mbase + v) * BATCH + b) * C + colv] = acc[v];
  }
}

// ---------------------------------------------------------------------------
// Per-node GEMM + bias + activation:
//   Out[n,b,o] = act( sum_i In[n,b,i]*Wn[n,0,i,o] + sum_i Cv[n,b,i]*Wn[n,1,i,o]
//                     + Bn[n,o] )
// M = batch (32 -> 2 m-tiles). Waves paired so both m-tiles of an o-tile sit
// in the same block (identical weight-row loads -> WGP$ reuse).
// grid = (N, ceil(Otiles/4)), block = 256.
// ---------------------------------------------------------------------------
__global__ void k_gemm_node(const float* __restrict__ In, const float* __restrict__ Cv,
                            const float* __restrict__ Wn, const float* __restrict__ Bn,
                            float* __restrict__ Out, int C, int O, int act) {
  const int wave = threadIdx.x >> 5;
  const int lane = threadIdx.x & 31;
  const int Otiles = O >> 4;
  const int otile = blockIdx.y * 4 + (wave >> 1);
  if (otile >= Otiles) return;
  const int mtile = wave & 1;
  const int n = blockIdx.x;
  const int m = lane & 15;
  const int kb = (lane < 16) ? 0 : 2;
  const int nn = lane & 15;
  const int bidx = mtile * 16 + m;   // batch row for A-operand
  const int ocol = otile * 16 + nn;
  const int Kpad = (C + 3) & ~3;

  v8f acc = {};
  for (int s = 0; s < 2; ++s) {
    const float* __restrict__ Xs = s ? Cv : In;
    const float* __restrict__ Arow = Xs + ((size_t)n * BATCH + bidx) * C;
    const float* __restrict__ Wb = Wn + ((size_t)n * 2 + s) * (size_t)C * O;
#pragma unroll 4
    for (int kk = 0; kk < Kpad; kk += 4) {
      const int k0 = kk + kb;
      const bool ok0 = (k0 < C), ok1 = (k0 + 1 < C);
      float a0 = Arow[ok0 ? k0 : 0];
      float a1 = Arow[ok1 ? (k0 + 1) : 0];
      float w0 = Wb[ok0 ? ((size_t)k0 * O + ocol) : 0];
      float w1 = Wb[ok1 ? ((size_t)(k0 + 1) * O + ocol) : 0];
      v2f av, bv;
      av.x = ok0 ? a0 : 0.f;
      av.y = ok1 ? a1 : 0.f;
      bv.x = ok0 ? w0 : 0.f;
      bv.y = ok1 ? w1 : 0.f;
      acc = wmma_f32(av, bv, acc);
    }
  }

  const float bias = Bn[(size_t)n * O + ocol];
  const int mbase = mtile * 16 + ((lane < 16) ? 0 : 8);
#pragma unroll
  for (int v = 0; v < 8; ++v) {
    float r = acc[v] + bias;
    r = act ? tanhf(r) : sigmoidf_(r);
    Out[((size_t)n * BATCH + (mbase + v)) * O + ocol] = r;
  }
}

// ---------------------------------------------------------------------------
// GRU combine: new = r*state + (1-r)*hc ; write seq / outputs
// ---------------------------------------------------------------------------
__global__ void k_gru(const float* __restrict__ zr, const float* __restrict__ hc,
                      float* __restrict__ st, float* __restrict__ h0seq,
                      float* __restrict__ dout, int layer, int t) {
  size_t idx = (size_t)blockIdx.x * blockDim.x + threadIdx.x;
  const size_t total = (size_t)N_NODES * BATCH * HID;
  if (idx >= total) return;
  int n = (int)(idx / (BATCH * HID));
  int rem = (int)(idx % (BATCH * HID));
  int b = rem / HID, o = rem % HID;

  float r = zr[((size_t)n * BATCH + b) * (2 * HID) + HID + o];
  float s = st[idx];
  float h = hc[idx];
  float nw = r * s + (1.0f - r) * h;
  st[idx] = nw;

  if (layer == 0) {
    h0seq[(size_t)t * total + idx] = nw;
  } else {
    dout[(((size_t)b * TSTEPS + t) * N_NODES + n) * HID + o] = nw;
  }
  if (t == TSTEPS - 1) {
    const size_t base = (size_t)BATCH * TSTEPS * N_NODES * HID +
                        (size_t)layer * BATCH * N_NODES * HID;
    dout[base + ((size_t)b * N_NODES + n) * HID + o] = nw;
  }
}

// ---------------------------------------------------------------------------
extern "C" void kernel_launch(void* const* d_in, const int* in_sizes, int n_in,
                              void* d_out, int out_size, void* d_ws, size_t ws_size,
                              hipStream_t stream) {
  const float* x   = (const float*)d_in[0];
  const float* ist = (const float*)d_in[1];
  const float* E   = (const float*)d_in[2];
  const float* gw0 = (const float*)d_in[3];
  const float* gb0 = (const float*)d_in[4];
  const float* uw0 = (const float*)d_in[5];
  const float* ub0 = (const float*)d_in[6];
  const float* gw1 = (const float*)d_in[7];
  const float* gb1 = (const float*)d_in[8];
  const float* uw1 = (const float*)d_in[9];
  const float* ub1 = (const float*)d_in[10];
  float* out = (float*)d_out;
  float* ws = (float*)d_ws;

  // ---- workspace carve (floats) ----
  size_t off = 0;
  auto take = [&](size_t nf) { float* p = ws + off; off += nf; return p; };
  float* Amat = take((size_t)N_NODES * N_NODES);                 // 4 MB
  float* Wg0  = take((size_t)N_NODES * 2 * 66 * 128);
  float* Bg0  = take((size_t)N_NODES * 128);
  float* Wu0  = take((size_t)N_NODES * 2 * 66 * 64);
  float* Bu0  = take((size_t)N_NODES * 64);
  float* Wg1  = take((size_t)N_NODES * 2 * 128 * 128);
  float* Bg1  = take((size_t)N_NODES * 128);
  float* Wu1  = take((size_t)N_NODES * 2 * 128 * 64);
  float* Bu1  = take((size_t)N_NODES * 64);
  float* st0  = take((size_t)N_NODES * BATCH * HID);
  float* st1  = take((size_t)N_NODES * BATCH * HID);
  float* inp  = take((size_t)N_NODES * BATCH * 128);             // reused as cand
  float* conv = take((size_t)N_NODES * BATCH * 128);
  float* zrb  = take((size_t)N_NODES * BATCH * 128);
  float* hcb  = take((size_t)N_NODES * BATCH * HID);
  float* h0sq = take((size_t)TSTEPS * N_NODES * BATCH * HID);

  auto g1 = [](size_t total) { return dim3((unsigned)((total + 255) / 256)); };

  // ---- one-time (per call) setup: graph + per-node weights ----
  k_scores<<<dim3(N_NODES), dim3(256), 0, stream>>>(E, Amat);
  k_mix<<<g1((size_t)N_NODES * 2 * 66 * 128), 256, 0, stream>>>(E, gw0, Wg0, 2 * 66 * 128);
  k_mix<<<g1((size_t)N_NODES * 128), 256, 0, stream>>>(E, gb0, Bg0, 128);
  k_mix<<<g1((size_t)N_NODES * 2 * 66 * 64), 256, 0, stream>>>(E, uw0, Wu0, 2 * 66 * 64);
  k_mix<<<g1((size_t)N_NODES * 64), 256, 0, stream>>>(E, ub0, Bu0, 64);
  k_mix<<<g1((size_t)N_NODES * 2 * 128 * 128), 256, 0, stream>>>(E, gw1, Wg1, 2 * 128 * 128);
  k_mix<<<g1((size_t)N_NODES * 128), 256, 0, stream>>>(E, gb1, Bg1, 128);
  k_mix<<<g1((size_t)N_NODES * 2 * 128 * 64), 256, 0, stream>>>(E, uw1, Wu1, 2 * 128 * 64);
  k_mix<<<g1((size_t)N_NODES * 64), 256, 0, stream>>>(E, ub1, Bu1, 64);

  const size_t stTotal = (size_t)N_NODES * BATCH * HID;

  for (int layer = 0; layer < 2; ++layer) {
    const int Cin = (layer == 0) ? DIN : HID;
    const int C = Cin + HID;                 // 66 or 128
    const int ctiles = (C + 15) / 16;
    const float* Wg = (layer == 0) ? Wg0 : Wg1;
    const float* Bg = (layer == 0) ? Bg0 : Bg1;
    const float* Wu = (layer == 0) ? Wu0 : Wu1;
    const float* Bu = (layer == 0) ? Bu0 : Bu1;
    float* st = (layer == 0) ? st0 : st1;
    const unsigned gyGate = ((128 / 16) + 3) / 4;   // 2
    const unsigned gyUpd  = ((64 / 16) + 3) / 4;    // 1

    k_load_state<<<g1(stTotal), 256, 0, stream>>>(ist, st, layer);

    for (int t = 0; t < TSTEPS; ++t) {
      const size_t nbc = (size_t)N_NODES * BATCH * C;
      // inp = [x_t | state]
      k_build<<<g1(nbc), 256, 0, stream>>>(x, h0sq, st, zrb, inp, layer, t, Cin, 0);
      // conv = A @ inp (per batch)
      k_gemm_graph<<<dim3(N_NODES / 128, ctiles, BATCH), 256, 0, stream>>>(Amat, inp, conv, C);
      // z_r = sigmoid(node-gemm)
      k_gemm_node<<<dim3(N_NODES, gyGate), 256, 0, stream>>>(inp, conv, Wg, Bg, zrb, C, 128, 0);
      // cand = [x_t | z*state]  (reuse inp buffer)
      k_build<<<g1(nbc), 256, 0, stream>>>(x, h0sq, st, zrb, inp, layer, t, Cin, 1);
      k_gemm_graph<<<dim3(N_NODES / 128, ctiles, BATCH), 256, 0, stream>>>(Amat, inp, conv, C);
      // hc = tanh(node-gemm)
      k_gemm_node<<<dim3(N_NODES, gyUpd), 256, 0, stream>>>(inp, conv, Wu, Bu, hcb, C, 64, 1);
      // state update + outputs
      k_gru<<<g1(stTotal), 256, 0, stream>>>(zrb, hcb, st, h0sq, out, layer, t);
    }
  }
}